// _AddShiftFallback_56831007260948
// MI455X (gfx1250) — compile-verified
//
#include <hip/hip_runtime.h>
#include <cstdint>

// Problem constants (fixed by the reference):
//   B=8, C_OUT=64, NK=11, H=W=64, GROUP_IN=4
#define NPIX     4096      // H*W
#define NCH      11        // nk
#define NCHP     12        // nk padded to multiple of WMMA K=4
#define NTHREADS 256       // 8 wave32 waves
#define NWAVES   8

typedef float v2f __attribute__((ext_vector_type(2)));
typedef float v8f __attribute__((ext_vector_type(8)));

// CDNA5 async global->LDS staging path (ASYNCcnt), gated on builtin presence.
#if defined(__has_builtin)
# if __has_builtin(__builtin_amdgcn_global_load_async_to_lds_b128) && \
     __has_builtin(__builtin_amdgcn_s_wait_asynccnt)
#  define USE_ASYNC_LDS 1
# endif
#endif
#ifndef USE_ASYNC_LDS
# define USE_ASYNC_LDS 0
#endif

// Builtin expects pointers to 16-byte int vectors (per hipcc diagnostic:
// "__attribute__((__vector_size__(4 * sizeof(int)))) int __device__ *").
typedef int v4i __attribute__((vector_size(4 * sizeof(int))));
typedef __attribute__((address_space(1))) v4i as1_v4i;   // global
typedef __attribute__((address_space(3))) v4i as3_v4i;   // LDS

// LDS layout (floats):
//   xs[12][4096]  staged x slab (ch 11 zero)  : 49152 floats
//   P [ 7][4096]  WMMA result planes          : 28672 floats @ +49152
//   Wl[16][12]    weight matrix (A operand)   :   192 floats @ +77824
// total = 312,064 bytes  (<= 320KB WGP LDS)
#define XS_OFF 0
#define P_OFF  49152
#define W_OFF  (49152 + 28672)
#define SMEM_FLOATS (49152 + 28672 + 192)

__global__ __launch_bounds__(NTHREADS)
void shiftsum_wmma_kernel(const float* __restrict__ x,
                          const int*   __restrict__ perm_h,
                          const int*   __restrict__ perm_v,
                          const int*   __restrict__ sel_id,
                          float*       __restrict__ out)
{
    extern __shared__ float smem[];
    float* xs = smem + XS_OFF;
    float* P  = smem + P_OFF;
    float* Wl = smem + W_OFF;

    const int tid = threadIdx.x;
    const int bid = blockIdx.x;      // bid = b*64 + r   (grid = 512)
    const int r   = bid & 63;

    // x slab for this (b, r): (b*704 + r*11)*4096 == bid * 11 * 4096
    const float* gsrc = x + (size_t)bid * (size_t)(NCH * NPIX);

    // ---------------- Phase 1a: stage 45056 floats (11 ch) into LDS ----------
#if USE_ASYNC_LDS
    {
        // Flat shared pointers are {aperture_hi32, lds_offset}; the low 32
        // bits are the exact AS3 byte address.
        const unsigned lbase = (unsigned)(uintptr_t)(void*)xs;
        for (int i = tid; i < (NCH * NPIX) / 4; i += NTHREADS) {
            __builtin_amdgcn_global_load_async_to_lds_b128(
                (as1_v4i*)(uintptr_t)(gsrc + (size_t)i * 4),
                (as3_v4i*)(uintptr_t)(lbase + (unsigned)i * 16u),
                0, 0);
        }
    }
#else
    for (int i = tid; i < (NCH * NPIX) / 4; i += NTHREADS) {
        const float4 v = ((const float4*)gsrc)[i];
        ((float4*)xs)[i] = v;
    }
#endif

    // Zero padding channel 11 (K-pad for WMMA; avoids NaN garbage in B).
    {
        float4 z = {0.f, 0.f, 0.f, 0.f};
        for (int i = tid; i < NPIX / 4; i += NTHREADS)
            ((float4*)(xs + NCH * NPIX))[i] = z;
    }

    // ---------------- Phase 1b: build weight matrix A = Wl[16][12] -----------
    // Rows 0..2 : lora1 shift-classes j=0..2 (class of position k is (r*11+k)%3)
    // Rows 3..5 : lora2 shift-classes j=0..2
    // Row  6    : small (sel_id counts);  rows 7..15 and col 11 stay zero.
    if (tid < 16) {
        #pragma unroll
        for (int q = 0; q < NCHP; ++q) Wl[tid * NCHP + q] = 0.f;
        if (tid < 3) {
            const int j = tid;
            for (int g = 0; g < 4; ++g)
                for (int k = 0; k < NCH; ++k)
                    if ((r * NCH + k) % 3 == j) {
                        const int c = perm_h[g * 704 + r * NCH + k] % NCH;
                        Wl[j * NCHP + c] += 1.f;
                    }
        } else if (tid < 6) {
            const int j = tid - 3;
            for (int g = 0; g < 4; ++g)
                for (int k = 0; k < NCH; ++k)
                    if ((r * NCH + k) % 3 == j) {
                        const int c = perm_v[g * 704 + r * NCH + k] % NCH;
                        Wl[tid * NCHP + c] += 1.f;
                    }
        } else if (tid == 6) {
            for (int g = 0; g < 4; ++g) {
                const int c = sel_id[g * 64 + r] % NCH;
                Wl[6 * NCHP + c] += 1.f;
            }
        }
    }

#if USE_ASYNC_LDS
    __builtin_amdgcn_s_wait_asynccnt(0);
#endif
    __syncthreads();

    // ---------------- Phase 2: planes P[7][4096] = A(16x12) @ xs(12x4096) ----
    // V_WMMA_F32_16X16X4_F32, K=12 in 3 k-steps; 16-pixel N-tiles, 256 tiles,
    // distributed across the 8 waves. EXEC is all-1s at every wmma.
    const int  lane = tid & 31;
    const int  wave = tid >> 5;
    const int  lrow = lane & 15;        // M row (A) / N pixel (B,D)
    const bool hi   = lane >= 16;

    // A fragments (constant across tiles). 32-bit 16x4 A layout:
    // lanes 0-15: {K=0,K=1}; lanes 16-31: {K=2,K=3} for row M=lane%16.
    v2f a0, a1, a2;
    {
        const int kb = hi ? 2 : 0;
        a0.x = Wl[lrow * NCHP + 0 + kb];  a0.y = Wl[lrow * NCHP + 0 + kb + 1];
        a1.x = Wl[lrow * NCHP + 4 + kb];  a1.y = Wl[lrow * NCHP + 4 + kb + 1];
        a2.x = Wl[lrow * NCHP + 8 + kb];  a2.y = Wl[lrow * NCHP + 8 + kb + 1];
    }

    for (int t = wave; t < NPIX / 16; t += NWAVES) {
        const int pix = t * 16 + lrow;
        const int cb  = hi ? 2 : 0;     // B 4x16 layout mirrors A K-split
        v2f b0, b1, b2;
        b0.x = xs[(cb + 0) * NPIX + pix];  b0.y = xs[(cb + 1) * NPIX + pix];
        b1.x = xs[(cb + 4) * NPIX + pix];  b1.y = xs[(cb + 5) * NPIX + pix];
        b2.x = xs[(cb + 8) * NPIX + pix];  b2.y = xs[(cb + 9) * NPIX + pix];

        v8f c = {0.f, 0.f, 0.f, 0.f, 0.f, 0.f, 0.f, 0.f};
        c = __builtin_amdgcn_wmma_f32_16x16x4_f32(false, a0, false, b0, (short)0, c, false, false);
        c = __builtin_amdgcn_wmma_f32_16x16x4_f32(false, a1, false, b1, (short)0, c, false, false);
        c = __builtin_amdgcn_wmma_f32_16x16x4_f32(false, a2, false, b2, (short)0, c, false, false);

        // D layout: VGPR v holds M=v (lanes 0-15) / M=v+8 (lanes 16-31).
        // Rows 0..6 are the useful planes -> lanes 0-15 store them.
        if (!hi) {
            #pragma unroll
            for (int v = 0; v < 7; ++v)
                P[v * NPIX + pix] = c[v];
        }
    }

    __syncthreads();

    // ---------------- Phase 3: circular-shift combine + global stores --------
    // lora1 = P0(h,w) + P1(h-1,w-2) + P2(h-2,w-1)
    // lora2 = P3(h,w) + P4(h-2,w-1) + P5(h-1,w-2)
    // small = P6(h,w)                       (all mod-64 wraps)
    const size_t obase = (size_t)bid * NPIX;
    float* o1 = out + obase;
    float* o2 = out + 2097152u + obase;
    float* o3 = out + 4194304u + obase;

    for (int p = tid; p < NPIX; p += NTHREADS) {
        const int h   = p >> 6, w = p & 63;
        const int hm1 = (h + 63) & 63, hm2 = (h + 62) & 63;
        const int wm1 = (w + 63) & 63, wm2 = (w + 62) & 63;
        const int p12 = (hm1 << 6) | wm2;   // (h-1, w-2)
        const int p21 = (hm2 << 6) | wm1;   // (h-2, w-1)
        o1[p] = P[0 * NPIX + p] + P[1 * NPIX + p12] + P[2 * NPIX + p21];
        o2[p] = P[3 * NPIX + p] + P[4 * NPIX + p21] + P[5 * NPIX + p12];
        o3[p] = P[6 * NPIX + p];
    }
}

extern "C" void kernel_launch(void* const* d_in, const int* in_sizes, int n_in,
                              void* d_out, int out_size, void* d_ws, size_t ws_size,
                              hipStream_t stream) {
    (void)in_sizes; (void)n_in; (void)out_size; (void)d_ws; (void)ws_size;
    const float* x      = (const float*)d_in[0];
    const int*   perm_h = (const int*)d_in[1];
    const int*   perm_v = (const int*)d_in[2];
    const int*   sel_id = (const int*)d_in[3];
    float*       out    = (float*)d_out;

    // One workgroup per (b, r): 8*64 = 512 blocks, 256 threads (8 wave32).
    // Dynamic LDS: 312,064 bytes (xs + planes + weights) per WGP.
    shiftsum_wmma_kernel<<<dim3(512), dim3(NTHREADS),
                           SMEM_FLOATS * sizeof(float), stream>>>(
        x, perm_h, perm_v, sel_id, out);
}